// FT_Conv_matrix_61426622267728
// MI455X (gfx1250) — compile-verified
//
#include <hip/hip_runtime.h>
#include <math.h>

typedef __attribute__((ext_vector_type(2))) float v2f;
typedef __attribute__((ext_vector_type(8))) float v8f;

#define TT 96          // time dim / modes
#define CC 64          // channel dim
#define BB 64          // batch
#define TC 6144        // TT*CC
#define PI_F 3.14159265358979323846f

// ---------------------------------------------------------------------------
// Stage 1:  A[t][o] = sum_f wf[f,o] * e^{-2*pi*i*f*t/96}   (complex, 96x96)
// block = t, thread = o (and doubles as f for twiddle fill)
// ---------------------------------------------------------------------------
__global__ __launch_bounds__(TT) void k_stageA(const float* __restrict__ wfr,
                                               const float* __restrict__ wfi,
                                               float* __restrict__ Ar,
                                               float* __restrict__ Ai) {
    __shared__ float twr[TT], twi[TT];
    const int t = blockIdx.x;
    const int o = threadIdx.x;
    {
        const int f = threadIdx.x;
        float s, c;
        sincosf(-2.0f * PI_F * (float)f * (float)t / (float)TT, &s, &c);
        twr[f] = c; twi[f] = s;
    }
    __syncthreads();
    float ar = 0.f, ai = 0.f;
    for (int f = 0; f < TT; ++f) {
        const float wr = wfr[f * TT + o];
        const float wi = wfi[f * TT + o];
        ar += wr * twr[f] - wi * twi[f];
        ai += wr * twi[f] + wi * twr[f];
    }
    Ar[t * TT + o] = ar;   // stored as Astore[t][o] = A(o,t)
    Ai[t * TT + o] = ai;
}

// ---------------------------------------------------------------------------
// Stage 2:  M[tau][t] = (1/96) * sum_o [cos(2pi*tau*o/96)*ReA(o,t)
//                                       - sin(2pi*tau*o/96)*ImA(o,t)]
// block = tau, thread = t (and doubles as o for twiddle fill)
// ---------------------------------------------------------------------------
__global__ __launch_bounds__(TT) void k_stageM(const float* __restrict__ Ar,
                                               const float* __restrict__ Ai,
                                               float* __restrict__ M) {
    __shared__ float cr[TT], sr[TT];
    const int tau = blockIdx.x;
    const int t   = threadIdx.x;
    {
        const int o = threadIdx.x;
        float s, c;
        sincosf(2.0f * PI_F * (float)tau * (float)o / (float)TT, &s, &c);
        cr[o] = c; sr[o] = s;
    }
    __syncthreads();
    float acc = 0.f;
    const float* art = Ar + t * TT;   // Astore[t][o] = A(o,t): contiguous in o
    const float* ait = Ai + t * TT;
    for (int o = 0; o < TT; ++o)
        acc += cr[o] * art[o] - sr[o] * ait[o];
    M[tau * TT + t] = acc * (1.0f / (float)TT);
}

// ---------------------------------------------------------------------------
// Main fused kernel: one block per batch b.
//   out[b,t,c] = (M @ x[b])[t,c] + relu(mid[b,t,c])
//   mid[b,t,c] = x_flat[b,t-1]*W1[t-1, t*64+c]  (1<=t<=94)
//              + x_flat[b,t  ]*W1[t  , t*64+c]  (t<=93)
// GEMM via v_wmma_f32_16x16x4_f32, fragments from LDS.
// ---------------------------------------------------------------------------
__global__ __launch_bounds__(256) void k_main(const float* __restrict__ x,
                                              const float* __restrict__ W1,
                                              const float* __restrict__ M,
                                              float* __restrict__ out) {
    __shared__ float Ml[TT * 97];   // M padded: row stride 97 (bank spread)
    __shared__ float Xl[TT * 65];   // x[b] padded: row stride 65

    const int b   = blockIdx.x;
    const int tid = threadIdx.x;

    // Stage M (9216 floats) and x[b] (6144 floats) into LDS.
    for (int i = tid; i < TT * TT; i += 256)
        Ml[(i / TT) * 97 + (i % TT)] = M[i];
    const float* xb = x + (size_t)b * TC;
    for (int i = tid; i < TC; i += 256)
        Xl[(i >> 6) * 65 + (i & 63)] = xb[i];

    // Prefetch the 48KB live band of weights_1 (rows 0..93, 128 floats each,
    // 4 cachelines per row) -> global_prefetch_b8, band stays L2/WGP$ hot.
    for (int i = tid; i < 94 * 4; i += 256) {
        const int row = i >> 2;
        __builtin_prefetch(W1 + (size_t)row * TC + row * 64 + (i & 3) * 32, 0, 0);
    }
    __syncthreads();

    const int wave = tid >> 5;
    const int lane = tid & 31;
    const int half = lane >> 4;   // lane group 0-15 vs 16-31
    const int r16  = lane & 15;

    // 24 output tiles (6 x 4 of 16x16) over 8 waves -> 3 tiles each.
    for (int s = 0; s < 3; ++s) {
        const int tile = wave + 8 * s;
        const int tm = tile >> 2;   // 0..5  (time-tile)
        const int tn = tile & 3;    // 0..3  (channel-tile)

        v8f acc = {};
        const int arow = (tm * 16 + r16) * 97;   // A: lane -> row of M
        const int bcol = tn * 16 + r16;          // B: lane -> col of x[b]

        // K = 96 in 24 steps of 4. A layout: lanes 0-15 hold K{0,1},
        // lanes 16-31 hold K{2,3}; B mirrored on the K rows.
        for (int kk = 0; kk < 24; ++kk) {
            const int k = kk * 4 + 2 * half;
            v2f a, bf;
            a.x  = Ml[arow + k];
            a.y  = Ml[arow + k + 1];
            bf.x = Xl[(k    ) * 65 + bcol];
            bf.y = Xl[(k + 1) * 65 + bcol];
            acc = __builtin_amdgcn_wmma_f32_16x16x4_f32(
                false, a, false, bf, (short)0, acc, false, false);
        }

        // Epilogue: C/D layout -> lane(0-15): row v, lane(16-31): row v+8.
        for (int v = 0; v < 8; ++v) {
            const int t = tm * 16 + v + 8 * half;
            const int c = tn * 16 + r16;
            const int j = t * 64 + c;            // flat column into W1 / out
            float mid = 0.f;
            if (t >= 1 && t <= 94)
                mid += Xl[((t - 1) >> 6) * 65 + ((t - 1) & 63)]
                     * W1[(size_t)(t - 1) * TC + j];
            if (t <= 93)
                mid += Xl[(t >> 6) * 65 + (t & 63)]
                     * W1[(size_t)t * TC + j];
            out[(size_t)b * TC + j] = acc[v] + fmaxf(mid, 0.f);
        }
    }
}

// ---------------------------------------------------------------------------
extern "C" void kernel_launch(void* const* d_in, const int* in_sizes, int n_in,
                              void* d_out, int out_size, void* d_ws, size_t ws_size,
                              hipStream_t stream) {
    (void)in_sizes; (void)n_in; (void)out_size; (void)ws_size;
    const float* x   = (const float*)d_in[0];   // (64, 96, 64)
    const float* W1  = (const float*)d_in[1];   // (6144, 6144)
    const float* wfr = (const float*)d_in[2];   // (96, 96)
    const float* wfi = (const float*)d_in[3];   // (96, 96)
    // d_in[4] = mask matrix: structure applied analytically, never read.

    float* ws = (float*)d_ws;
    float* Ar = ws;                 // 9216 floats
    float* Ai = ws + TT * TT;       // 9216 floats
    float* Mm = ws + 2 * TT * TT;   // 9216 floats

    k_stageA<<<TT, TT, 0, stream>>>(wfr, wfi, Ar, Ai);
    k_stageM<<<TT, TT, 0, stream>>>(Ar, Ai, Mm);
    k_main<<<BB, 256, 0, stream>>>(x, W1, Mm, (float*)d_out);
}